// GCNLinkPredictor_11742440587907
// MI455X (gfx1250) — compile-verified
//
#include <hip/hip_runtime.h>
#include <hip/hip_bf16.h>

typedef __attribute__((ext_vector_type(2))) float v2f;
typedef __attribute__((ext_vector_type(8))) float v8f;

#define DDIM     128
#define NTOP     100
#define NROWPAD  112        // ceil(100/16)*16
#define TIE_CAP  131072

// ---------------------------------------------------------------------------
// K1: init workspace (degrees, row map, histograms, counters)
// ---------------------------------------------------------------------------
__global__ void k_init(unsigned* __restrict__ deg, int* __restrict__ rowMap,
                       unsigned* __restrict__ hist1, unsigned* __restrict__ hist2,
                       unsigned* __restrict__ counters, int N) {
    int i = blockIdx.x * blockDim.x + threadIdx.x;
    if (i < N)      { deg[i] = 0u; rowMap[i] = -1; }
    if (i < 65536)  { hist1[i] = 0u; hist2[i] = 0u; }
    if (i < 16)     { counters[i] = 0u; }
}

// ---------------------------------------------------------------------------
// K2: degree counts over source nodes
// ---------------------------------------------------------------------------
__global__ void k_degrees(const int* __restrict__ e0, unsigned* __restrict__ deg, int E) {
    int i = blockIdx.x * blockDim.x + threadIdx.x;
    if (i < E) atomicAdd(&deg[e0[i]], 1u);
}

// ---------------------------------------------------------------------------
// K3: top-100 nodes by degree (ties -> lower index first, matching lax.top_k)
// rank(i) = #{j : deg[j] > deg[i]} + #{j < i : deg[j] == deg[i]}
// ---------------------------------------------------------------------------
__global__ void k_top100(const unsigned* __restrict__ deg, int* __restrict__ topNodes,
                         int* __restrict__ rowMap, int N) {
    int i = blockIdx.x * blockDim.x + threadIdx.x;
    if (i >= N) return;
    unsigned di = deg[i];
    int rank = 0;
    for (int j = 0; j < N; ++j) {
        unsigned dj = deg[j];
        rank += (dj > di) || (dj == di && j < i);
    }
    if (rank < NTOP) { topNodes[rank] = i; rowMap[i] = rank; }
}

// ---------------------------------------------------------------------------
// K4: gather A = z[topNodes] into padded [112][128], zeros past row 99
// ---------------------------------------------------------------------------
__global__ void k_gather(const float* __restrict__ z, const int* __restrict__ topNodes,
                         float* __restrict__ Apad) {
    int i = blockIdx.x * blockDim.x + threadIdx.x;
    if (i >= NROWPAD * DDIM) return;
    int r = i / DDIM, d = i - r * DDIM;
    Apad[i] = (r < NTOP) ? z[(size_t)topNodes[r] * DDIM + d] : 0.0f;
}

// ---------------------------------------------------------------------------
// K5: WMMA GEMM  scores[112][N] = sigmoid(Apad @ z^T) + 1   (0 for pad rows)
// One wave per 16x16 output tile; V_WMMA_F32_16X16X4_F32, K chained 128/4=32.
// f32 16x4 A-frag: lane m=lane&15, half=lane>>4: {A[m][k+2h], A[m][k+2h+1]}
// f32 4x16 B-frag: lane n=lane&15, half=lane>>4: {B[k+2h][n], B[k+2h+1][n]}
//   with B = z^T  =>  {z[col0+n][k+2h], z[col0+n][k+2h+1]}  (contiguous!)
// C/D 16x16 f32: element (m,n) in lane n+(m>=8)*16, vgpr m&7.
// ---------------------------------------------------------------------------
__global__ void k_gemm_scores(const float* __restrict__ Apad, const float* __restrict__ z,
                              float* __restrict__ scores, int N) {
    const int wavesPerBlock = blockDim.x >> 5;
    int wave = blockIdx.x * wavesPerBlock + (threadIdx.x >> 5);
    int lane = threadIdx.x & 31;
    int colTiles = N >> 4;
    int tileRow = wave / colTiles;            // wave-uniform
    int tileCol = wave - tileRow * colTiles;
    if (tileRow >= (NROWPAD >> 4)) return;    // uniform across wave
    int row0 = tileRow << 4;
    int col0 = tileCol << 4;
    int hl = lane >> 4;       // 0 or 1
    int ml = lane & 15;

    const float* aptr = Apad + (size_t)(row0 + ml) * DDIM + hl * 2;
    const float* bptr = z    + (size_t)(col0 + ml) * DDIM + hl * 2;

    v8f acc = {};
#pragma unroll
    for (int kk = 0; kk < DDIM; kk += 4) {
        v2f a = *(const v2f*)(aptr + kk);
        v2f b = *(const v2f*)(bptr + kk);
        acc = __builtin_amdgcn_wmma_f32_16x16x4_f32(
            /*neg_a=*/false, a, /*neg_b=*/false, b,
            /*c_mod=*/(short)0, acc, /*reuse_a=*/false, /*reuse_b=*/false);
    }

#pragma unroll
    for (int v = 0; v < 8; ++v) {
        int m  = v + hl * 8;
        int gr = row0 + m;
        int gc = col0 + ml;
        float s   = acc[v];
        float val = (gr < NTOP) ? (1.0f + 1.0f / (1.0f + expf(-s))) : 0.0f;
        scores[(size_t)gr * N + gc] = val;
    }
}

// ---------------------------------------------------------------------------
// K6: zero scores at existing edges whose source is a top node (-1+1 == 0)
// ---------------------------------------------------------------------------
__global__ void k_mask(const int* __restrict__ e0, const int* __restrict__ e1,
                       const int* __restrict__ rowMap, float* __restrict__ scores,
                       int E, int N) {
    int i = blockIdx.x * blockDim.x + threadIdx.x;
    if (i >= E) return;
    int r = rowMap[e0[i]];
    if (r >= 0) scores[(size_t)r * N + e1[i]] = 0.0f;
}

// ---------------------------------------------------------------------------
// K7/K9: radix histograms over positive-float bit patterns (monotone)
// ---------------------------------------------------------------------------
__global__ void k_hist1(const float* __restrict__ scores, unsigned* __restrict__ hist1, int total) {
    int i = blockIdx.x * blockDim.x + threadIdx.x;
    if (i >= total) return;
    unsigned u = __float_as_uint(scores[i]);
    atomicAdd(&hist1[u >> 16], 1u);
}

__global__ void k_findbin1(const unsigned* __restrict__ hist1, unsigned* __restrict__ counters, int k) {
    if (blockIdx.x != 0 || threadIdx.x != 0) return;
    unsigned cum = 0;
    for (int b = 65535; b >= 0; --b) {
        unsigned h = hist1[b];
        if (cum + h >= (unsigned)k) { counters[2] = (unsigned)b; counters[3] = cum; return; }
        cum += h;
    }
    counters[2] = 0u; counters[3] = cum;
}

__global__ void k_hist2(const float* __restrict__ scores, const unsigned* __restrict__ counters,
                        unsigned* __restrict__ hist2, int total) {
    int i = blockIdx.x * blockDim.x + threadIdx.x;
    if (i >= total) return;
    unsigned u = __float_as_uint(scores[i]);
    if ((u >> 16) == counters[2]) atomicAdd(&hist2[u & 0xFFFFu], 1u);
}

__global__ void k_findthr(const unsigned* __restrict__ hist2, unsigned* __restrict__ counters, int k) {
    if (blockIdx.x != 0 || threadIdx.x != 0) return;
    unsigned B1 = counters[2], above = counters[3];
    unsigned k2 = (unsigned)k - above;       // >= 1 by construction
    unsigned cum = 0; int L = 0;
    for (int b = 65535; b >= 0; --b) {
        unsigned h = hist2[b];
        if (cum + h >= k2) { L = b; break; }
        cum += h;
    }
    counters[4] = (B1 << 16) | (unsigned)L;  // exact 32-bit threshold T
    counters[5] = above + cum;               // A = count strictly above T (< k)
}

// ---------------------------------------------------------------------------
// K11: compact candidates (> T) and ties (== T); store ORIGINAL flat index
// ---------------------------------------------------------------------------
__global__ void k_compact(const float* __restrict__ scores, const int* __restrict__ topNodes,
                          unsigned* __restrict__ counters, unsigned* __restrict__ candU,
                          int* __restrict__ candI, int* __restrict__ tieI, int total, int N) {
    int i = blockIdx.x * blockDim.x + threadIdx.x;
    if (i >= total) return;
    unsigned u = __float_as_uint(scores[i]);
    unsigned T = counters[4];
    if (u > T) {
        unsigned p = atomicAdd(&counters[0], 1u);
        int r = i / N, c = i - r * N;
        candU[p] = u;
        candI[p] = topNodes[r] * N + c;
    } else if (u == T) {
        unsigned p = atomicAdd(&counters[1], 1u);
        if (p < (unsigned)TIE_CAP) {
            int r = i / N, c = i - r * N;
            tieI[p] = topNodes[r] * N + c;
        }
    }
}

// ---------------------------------------------------------------------------
// K12: rank above-threshold candidates (value desc, idx asc) and emit.
// out layout: [rows k | cols k | cols k | rows k | w k | w k]  (all floats)
// ---------------------------------------------------------------------------
__global__ void k_rank(const unsigned* __restrict__ candU, const int* __restrict__ candI,
                       const unsigned* __restrict__ counters, float* __restrict__ out,
                       int k, int N) {
    __shared__ unsigned sU[256];
    __shared__ int      sI[256];
    int i = blockIdx.x * blockDim.x + threadIdx.x;
    unsigned A = counters[0];
    bool active = (i < (int)A);
    unsigned ui = active ? candU[i] : 0u;
    int      xi = active ? candI[i] : 0;
    int rank = 0;
    for (unsigned base = 0; base < A; base += 256u) {
        unsigned j = base + threadIdx.x;
        sU[threadIdx.x] = (j < A) ? candU[j] : 0u;
        sI[threadIdx.x] = (j < A) ? candI[j] : 0x7FFFFFFF;
        __syncthreads();
        unsigned lim = A - base; if (lim > 256u) lim = 256u;
        for (unsigned t = 0; t < lim; ++t) {
            unsigned uj = sU[t]; int xj = sI[t];
            rank += ((uj > ui) || (uj == ui && xj < xi)) ? 1 : 0;
        }
        __syncthreads();
    }
    if (!active) return;
    int r = xi / N, c = xi - r * N;
    float w = __uint_as_float(ui) - 1.0f;
    out[rank]          = (float)r;
    out[k + rank]      = (float)c;
    out[2*k + rank]    = (float)c;
    out[3*k + rank]    = (float)r;
    out[4*k + rank]    = w;
    out[5*k + rank]    = w;
}

// ---------------------------------------------------------------------------
// K13: fill positions A..k-1 with ties at T, ascending original flat index
// ---------------------------------------------------------------------------
__global__ void k_ties(const int* __restrict__ tieI, const unsigned* __restrict__ counters,
                       float* __restrict__ out, int k, int N) {
    int i = blockIdx.x * blockDim.x + threadIdx.x;
    unsigned A  = counters[0];
    unsigned tc = counters[1]; if (tc > (unsigned)TIE_CAP) tc = TIE_CAP;
    if (i >= (int)tc) return;
    int xi = tieI[i];
    int rank = 0;
    for (unsigned j = 0; j < tc; ++j) rank += (tieI[j] < xi) ? 1 : 0;
    int pos = (int)A + rank;
    if (pos >= k) return;
    unsigned T = counters[4];
    int r = xi / N, c = xi - r * N;
    float w = __uint_as_float(T) - 1.0f;
    out[pos]          = (float)r;
    out[k + pos]      = (float)c;
    out[2*k + pos]    = (float)c;
    out[3*k + pos]    = (float)r;
    out[4*k + pos]    = w;
    out[5*k + pos]    = w;
}

// ---------------------------------------------------------------------------
extern "C" void kernel_launch(void* const* d_in, const int* in_sizes, int n_in,
                              void* d_out, int out_size, void* d_ws, size_t ws_size,
                              hipStream_t stream) {
    const float* z  = (const float*)d_in[0];
    const int*   e0 = (const int*)d_in[1];
    const int    N  = in_sizes[0] / DDIM;
    const int    E  = in_sizes[1] / 2;
    const int*   e1 = e0 + E;
    float* out = (float*)d_out;
    const int k = out_size / 6;           // k recovered from output shape
    if (k <= 0 || N <= 0) return;

    // ---- carve workspace (256B aligned slabs) ----
    size_t off = 0;
    auto carve = [&](size_t bytes) -> void* {
        off = (off + 255) & ~(size_t)255;
        void* p = (char*)d_ws + off;
        off += bytes;
        return p;
    };
    unsigned* deg      = (unsigned*)carve((size_t)N * 4);
    int*      rowMap   = (int*)carve((size_t)N * 4);
    int*      topNodes = (int*)carve(128 * 4);
    unsigned* counters = (unsigned*)carve(16 * 4);
    unsigned* hist1    = (unsigned*)carve(65536 * 4);
    unsigned* hist2    = (unsigned*)carve(65536 * 4);
    float*    Apad     = (float*)carve((size_t)NROWPAD * DDIM * 4);
    float*    scores   = (float*)carve((size_t)NROWPAD * N * 4);
    unsigned* candU    = (unsigned*)carve((size_t)(k + 64) * 4);
    int*      candI    = (int*)carve((size_t)(k + 64) * 4);
    int*      tieI     = (int*)carve((size_t)TIE_CAP * 4);
    (void)ws_size; (void)n_in;

    const int B = 256;
    auto blk = [](long long t, int b) { return (unsigned)((t + b - 1) / b); };

    const int total  = NTOP * N;                 // scored entries (rows 0..99)
    const int initT  = (N > 65536) ? N : 65536;

    k_init     <<<blk(initT, B), B, 0, stream>>>(deg, rowMap, hist1, hist2, counters, N);
    k_degrees  <<<blk(E, B),     B, 0, stream>>>(e0, deg, E);
    k_top100   <<<blk(N, B),     B, 0, stream>>>(deg, topNodes, rowMap, N);
    k_gather   <<<blk(NROWPAD * DDIM, B), B, 0, stream>>>(z, topNodes, Apad);

    // GEMM: one wave per 16x16 tile; 8 waves/block (wave32)
    {
        long long tiles = (long long)(NROWPAD / 16) * (N / 16);
        unsigned blocks = blk(tiles * 32, B);
        k_gemm_scores<<<blocks, B, 0, stream>>>(Apad, z, scores, N);
    }

    k_mask     <<<blk(E, B),     B, 0, stream>>>(e0, e1, rowMap, scores, E, N);
    k_hist1    <<<blk(total, B), B, 0, stream>>>(scores, hist1, total);
    k_findbin1 <<<1, 32, 0, stream>>>(hist1, counters, k);
    k_hist2    <<<blk(total, B), B, 0, stream>>>(scores, counters, hist2, total);
    k_findthr  <<<1, 32, 0, stream>>>(hist2, counters, k);
    k_compact  <<<blk(total, B), B, 0, stream>>>(scores, topNodes, counters, candU, candI, tieI, total, N);
    k_rank     <<<blk(k, B),     B, 0, stream>>>(candU, candI, counters, out, k, N);
    k_ties     <<<blk(TIE_CAP, B), B, 0, stream>>>(tieI, counters, out, k, N);
}